// rbfn_61065845014989
// MI455X (gfx1250) — compile-verified
//
#include <hip/hip_runtime.h>
#include <math.h>

typedef __attribute__((ext_vector_type(2))) float v2f;
typedef __attribute__((ext_vector_type(8))) float v8f;

#define DELTA 0.1f
#define N_TOT 4096
#define D_DIM 256
#define C_DIM 512
#define U_DIM 256

// ---------------------------------------------------------------------------
// Prep: dst[row][d] = src[row][d] / norm[d];  sq[row] = sum_d dst[row][d]^2
// One 256-thread block (8 waves) per row; D_DIM == 256.
// ---------------------------------------------------------------------------
__global__ void rbfn_prep_normalize(const float* __restrict__ src,
                                    const float* __restrict__ norm,
                                    float* __restrict__ dst,
                                    float* __restrict__ sq) {
  const int row = blockIdx.x;
  const int d   = threadIdx.x;          // 0..255
  const float v = src[row * D_DIM + d] / norm[d];
  dst[row * D_DIM + d] = v;

  float s = v * v;
  #pragma unroll
  for (int off = 16; off > 0; off >>= 1)
    s += __shfl_down(s, off, 32);

  __shared__ float partial[8];
  const int lane = threadIdx.x & 31;
  const int wv   = threadIdx.x >> 5;
  if (lane == 0) partial[wv] = s;
  __syncthreads();
  if (threadIdx.x == 0) {
    float t = 0.f;
    #pragma unroll
    for (int i = 0; i < 8; ++i) t += partial[i];
    sq[row] = t;
  }
}

// ---------------------------------------------------------------------------
// GEMM1 + RBF epilogue, dual-accumulator (16x32 output per wave):
//   G = xn @ cn^T   (K = D_DIM) via V_WMMA_F32_16X16X4_F32
//   act[n,c] = exp(-DELTA * sqrt(xsq[n] + csq[c] - 2*G))
// blockDim = (32,4): 4 waves cover a 16x128 strip; gridDim = (N/16, C/128).
// ---------------------------------------------------------------------------
__global__ void rbfn_gemm1_rbf(const float* __restrict__ xn,
                               const float* __restrict__ cn,
                               const float* __restrict__ xsq,
                               const float* __restrict__ csq,
                               float* __restrict__ act) {
  const int lane = threadIdx.x;            // 0..31
  const int half = lane >> 4;              // 0 or 1
  const int m    = lane & 15;
  const int n0   = blockIdx.x * 16;
  const int c0   = (blockIdx.y * 4 + threadIdx.y) * 32;

  const float* __restrict__ arow  = xn + (size_t)(n0 + m) * D_DIM;
  const float* __restrict__ brow0 = cn + (size_t)(c0 + m) * D_DIM;
  const float* __restrict__ brow1 = cn + (size_t)(c0 + 16 + m) * D_DIM;

  v8f acc0 = {};
  v8f acc1 = {};
  #pragma unroll 4
  for (int k = 0; k < D_DIM; k += 4) {
    // A 16x4 f32: v0 holds K = 2*half, v1 holds K = 2*half+1 (per-lane pair)
    v2f a  = *(const v2f*)(arow  + k + 2 * half);
    // B 4x16 f32: cn row-major read as cn^T fragment; same per-lane indexing
    v2f b0 = *(const v2f*)(brow0 + k + 2 * half);
    v2f b1 = *(const v2f*)(brow1 + k + 2 * half);
    acc0 = __builtin_amdgcn_wmma_f32_16x16x4_f32(
        false, a, false, b0, (short)0, acc0, false, false);
    acc1 = __builtin_amdgcn_wmma_f32_16x16x4_f32(
        false, a, false, b1, (short)0, acc1, false, false);
  }

  const float cs0 = csq[c0 + m];
  const float cs1 = csq[c0 + 16 + m];
  #pragma unroll
  for (int r = 0; r < 8; ++r) {
    const int n = n0 + r + 8 * half;       // C/D layout: lanes16-31 -> M = r+8
    const float xs = xsq[n];
    float d2a = xs + cs0 - 2.0f * acc0[r];
    float d2b = xs + cs1 - 2.0f * acc1[r];
    d2a = d2a > 0.0f ? d2a : 0.0f;
    d2b = d2b > 0.0f ? d2b : 0.0f;
    act[(size_t)n * C_DIM + (c0 + m)]      = expf(-DELTA * sqrtf(d2a));
    act[(size_t)n * C_DIM + (c0 + 16 + m)] = expf(-DELTA * sqrtf(d2b));
  }
}

// ---------------------------------------------------------------------------
// GEMM2: out = act @ kernel   (K = C_DIM) via V_WMMA_F32_16X16X4_F32
// Dual accumulator: 16x32 output per wave; kernel[C,U] is K x N row-major,
// so B loads are lane-coalesced. gridDim = (N/16, U/128).
// ---------------------------------------------------------------------------
__global__ void rbfn_gemm2(const float* __restrict__ act,
                           const float* __restrict__ kern,
                           float* __restrict__ out) {
  const int lane = threadIdx.x;
  const int half = lane >> 4;
  const int m    = lane & 15;
  const int n0   = blockIdx.x * 16;
  const int u0   = (blockIdx.y * 4 + threadIdx.y) * 32;

  const float* __restrict__ arow = act + (size_t)(n0 + m) * C_DIM;

  v8f acc0 = {};
  v8f acc1 = {};
  #pragma unroll 4
  for (int k = 0; k < C_DIM; k += 4) {
    v2f a = *(const v2f*)(arow + k + 2 * half);
    const float* kr0 = kern + (size_t)(k + 2 * half) * U_DIM;
    const float* kr1 = kr0 + U_DIM;
    v2f b0, b1;
    b0.x = kr0[u0 + m];       b0.y = kr1[u0 + m];
    b1.x = kr0[u0 + 16 + m];  b1.y = kr1[u0 + 16 + m];
    acc0 = __builtin_amdgcn_wmma_f32_16x16x4_f32(
        false, a, false, b0, (short)0, acc0, false, false);
    acc1 = __builtin_amdgcn_wmma_f32_16x16x4_f32(
        false, a, false, b1, (short)0, acc1, false, false);
  }

  #pragma unroll
  for (int r = 0; r < 8; ++r) {
    const int n = n0 + r + 8 * half;
    out[(size_t)n * U_DIM + u0 + m]      = acc0[r];
    out[(size_t)n * U_DIM + u0 + 16 + m] = acc1[r];
  }
}

// ---------------------------------------------------------------------------
extern "C" void kernel_launch(void* const* d_in, const int* in_sizes, int n_in,
                              void* d_out, int out_size, void* d_ws, size_t ws_size,
                              hipStream_t stream) {
  const float* x       = (const float*)d_in[0];   // [4096, 256]
  const float* centers = (const float*)d_in[1];   // [512, 256]
  const float* norm    = (const float*)d_in[2];   // [256]
  const float* kern    = (const float*)d_in[3];   // [512, 256]
  float* out = (float*)d_out;                     // [4096, 256]

  // Workspace layout (floats): xn | cn | xsq | csq | act  (~12.5 MB)
  float* ws  = (float*)d_ws;
  float* xn  = ws;                                // 4096*256
  float* cn  = xn  + (size_t)N_TOT * D_DIM;       // 512*256
  float* xsq = cn  + (size_t)C_DIM * D_DIM;       // 4096
  float* csq = xsq + N_TOT;                       // 512
  float* act = csq + C_DIM;                       // 4096*512

  rbfn_prep_normalize<<<N_TOT, D_DIM, 0, stream>>>(x, norm, xn, xsq);
  rbfn_prep_normalize<<<C_DIM, D_DIM, 0, stream>>>(centers, norm, cn, csq);

  dim3 blk(32, 4);
  rbfn_gemm1_rbf<<<dim3(N_TOT / 16, C_DIM / 128), blk, 0, stream>>>(xn, cn, xsq, csq, act);
  rbfn_gemm2  <<<dim3(N_TOT / 16, U_DIM / 128), blk, 0, stream>>>(act, kern, out);
}